// TrainiumMatMulFP32_38276748542447
// MI455X (gfx1250) — compile-verified
//
#include <hip/hip_runtime.h>
#include <math.h>

typedef float v2f __attribute__((ext_vector_type(2)));
typedef float v4f __attribute__((ext_vector_type(4)));
typedef float v8f __attribute__((ext_vector_type(8)));

#define BM 128
#define BN 128
#define BK 16
#define LDA 20    // padded A tile row stride (floats): conflict-free frag loads, 16B-aligned fills
#define LDB 144   // padded B tile row stride (floats): halves land 32 banks apart

__global__ __launch_bounds__(256)
void fp32_gemm_wmma(const float* __restrict__ A, const float* __restrict__ B,
                    float* __restrict__ C, int M, int N, int K)
{
    __shared__ __align__(16) float As[2][BM * LDA];
    __shared__ __align__(16) float Bs[2][BK * LDB];

    const int tid  = threadIdx.x;
    const int wave = tid >> 5;
    const int lane = tid & 31;
    const int half = lane >> 4;   // 0: lanes 0-15, 1: lanes 16-31
    const int lr   = lane & 15;

    const int blockM = blockIdx.y * BM;
    const int blockN = blockIdx.x * BN;

    // wave sub-tile: 64 rows x 32 cols (4x2 tiles of 16x16)
    const int wm = (wave >> 2) * 64;   // 0 or 64
    const int wn = (wave & 3) * 32;    // 0,32,64,96

    // ---- global->LDS staging maps (two float4 each for A and B per thread) ----
    const int ar0 = tid >> 2;              // 0..63
    const int ar1 = ar0 + 64;              // 64..127
    const int ac  = (tid & 3) * 4;         // 0,4,8,12
    const int br0 = tid >> 5;              // 0..7
    const int br1 = br0 + 8;               // 8..15
    const int bc  = (tid & 31) * 4;        // 0..124

    const long aoff0 = (long)(blockM + ar0) * K + ac;
    const long aoff1 = (long)(blockM + ar1) * K + ac;
    const long boff0 = (long)br0 * N + blockN + bc;
    const long boff1 = (long)br1 * N + blockN + bc;

    v8f acc[4][2] = {};   // 64 VGPRs of accumulators

    const int KT = K / BK;

    // prologue: fill buffer 0
    {
        v4f ra0 = *(const v4f*)&A[aoff0];
        v4f ra1 = *(const v4f*)&A[aoff1];
        v4f rb0 = *(const v4f*)&B[boff0];
        v4f rb1 = *(const v4f*)&B[boff1];
        *(v4f*)&As[0][ar0 * LDA + ac] = ra0;
        *(v4f*)&As[0][ar1 * LDA + ac] = ra1;
        *(v4f*)&Bs[0][br0 * LDB + bc] = rb0;
        *(v4f*)&Bs[0][br1 * LDB + bc] = rb1;
    }
    __syncthreads();

    for (int kt = 0; kt < KT; ++kt) {
        const int buf = kt & 1;

        // issue next tile's global loads early so they fly under the WMMA work
        v4f ra0, ra1, rb0, rb1;
        const bool more = (kt + 1) < KT;
        if (more) {
            const int k0 = (kt + 1) * BK;
            ra0 = *(const v4f*)&A[aoff0 + k0];
            ra1 = *(const v4f*)&A[aoff1 + k0];
            rb0 = *(const v4f*)&B[boff0 + (long)k0 * N];
            rb1 = *(const v4f*)&B[boff1 + (long)k0 * N];
        }

        // ---- compute on buf: BK=16 -> four k-steps of 4 ----
        #pragma unroll
        for (int ks = 0; ks < BK; ks += 4) {
            v2f af[4];
            v2f bf[2];
            #pragma unroll
            for (int tm = 0; tm < 4; ++tm) {
                // A frag: lane half selects K pair (ks,ks+1) vs (ks+2,ks+3)
                af[tm] = *(const v2f*)&As[buf][(wm + tm * 16 + lr) * LDA + ks + 2 * half];
            }
            #pragma unroll
            for (int tn = 0; tn < 2; ++tn) {
                const int col = wn + tn * 16 + lr;
                bf[tn].x = Bs[buf][(ks + 2 * half    ) * LDB + col];
                bf[tn].y = Bs[buf][(ks + 2 * half + 1) * LDB + col];
            }
            #pragma unroll
            for (int tm = 0; tm < 4; ++tm)
                #pragma unroll
                for (int tn = 0; tn < 2; ++tn)
                    acc[tm][tn] = __builtin_amdgcn_wmma_f32_16x16x4_f32(
                        /*neg_a=*/false, af[tm],
                        /*neg_b=*/false, bf[tn],
                        /*c_mod=*/(short)0, acc[tm][tn],
                        /*reuse_a=*/false, /*reuse_b=*/false);
        }

        // stage next tile into the other buffer
        if (more) {
            *(v4f*)&As[buf ^ 1][ar0 * LDA + ac] = ra0;
            *(v4f*)&As[buf ^ 1][ar1 * LDA + ac] = ra1;
            *(v4f*)&Bs[buf ^ 1][br0 * LDB + bc] = rb0;
            *(v4f*)&Bs[buf ^ 1][br1 * LDB + bc] = rb1;
        }
        __syncthreads();
    }

    // ---- epilogue: C/D layout: VGPR r -> row r (lanes 0-15) / row r+8 (lanes 16-31) ----
    #pragma unroll
    for (int tm = 0; tm < 4; ++tm) {
        #pragma unroll
        for (int tn = 0; tn < 2; ++tn) {
            const int col = blockN + wn + tn * 16 + lr;
            #pragma unroll
            for (int r = 0; r < 8; ++r) {
                const int row = blockM + wm + tm * 16 + half * 8 + r;
                C[(long)row * N + col] = acc[tm][tn][r];
            }
        }
    }
}

extern "C" void kernel_launch(void* const* d_in, const int* in_sizes, int n_in,
                              void* d_out, int out_size, void* d_ws, size_t ws_size,
                              hipStream_t stream) {
    const float* A = (const float*)d_in[0];
    const float* B = (const float*)d_in[1];
    float* C = (float*)d_out;

    // Recover dims: in_sizes[0]=M*K, in_sizes[1]=K*N, out_size=M*N
    const long mk = (long)in_sizes[0];
    const long kn = (long)in_sizes[1];
    const long mn = (long)out_size;
    const int K = (int)llround(sqrt((double)mk * (double)kn / (double)mn));
    const int M = (int)(mk / K);
    const int N = (int)(kn / K);

    dim3 grid(N / BN, M / BM);
    fp32_gemm_wmma<<<grid, 256, 0, stream>>>(A, B, C, M, N, K);
}